// RGCNBasisLayer_5446018531333
// MI455X (gfx1250) — compile-verified
//
#include <hip/hip_runtime.h>

typedef float v2f __attribute__((ext_vector_type(2)));
typedef float v8f __attribute__((ext_vector_type(8)));

#define D 64
#define BASES 4
#define TILES_PER_BLOCK 8   // 8 waves of 32 per block, one 16-row tile each

// ---------------------------------------------------------------------------
// Packed B layout for WMMA f32 16x16x4:
//   Wp[(k>>1)*2*D + col*2 + (k&1)] = W[k*D + col]
// so a lane's B fragment (K-pair kb,kb+1 at column col) is one aligned b64:
//   *(v2f*)(Wp + (kb>>1)*2*D + col*2)
// ---------------------------------------------------------------------------

// K1: combined basis weights, written directly in packed layout.
__global__ void rgcn_basis_combine(const float* __restrict__ weight,
                                   const float* __restrict__ w_comp,
                                   float* __restrict__ Wp, int R) {
  int idx = blockIdx.x * blockDim.x + threadIdx.x;
  if (idx >= R * D * D) return;
  int r   = idx / (D * D);
  int rem = idx % (D * D);
  int k   = rem / D;
  int col = rem % D;
  float s = 0.f;
#pragma unroll
  for (int b = 0; b < BASES; ++b)
    s = fmaf(w_comp[r * BASES + b], weight[b * D * D + k * D + col], s);
  Wp[(size_t)r * D * D + (k >> 1) * 2 * D + col * 2 + (k & 1)] = s;
}

// K1b: pack the self-loop weight (an input) into the same layout.
__global__ void rgcn_pack_w64(const float* __restrict__ Wsrc,
                              float* __restrict__ Wdst) {
  int idx = blockIdx.x * blockDim.x + threadIdx.x;
  if (idx >= D * D) return;
  int k = idx / D, col = idx % D;
  Wdst[(k >> 1) * 2 * D + col * 2 + (k & 1)] = Wsrc[k * D + col];
}

// f32 WMMA: D(16x16) = A(16x4) * B(4x16) + C  (wave32)
__device__ __forceinline__ v8f wmma_f32_16x16x4(v2f a, v2f b, v8f c) {
  return __builtin_amdgcn_wmma_f32_16x16x4_f32(false, a, false, b,
                                               (short)0, c, false, false);
}

// ---------------------------------------------------------------------------
// K2: xr[r] = feat @ W[r]   ([R,N,64]), one wave per 16x64 output slab.
// A-frag (ISA 7.12.2, 32-bit A 16x4): lanes 0..15 hold K={4kk,4kk+1},
// lanes 16..31 hold K={4kk+2,4kk+3}, rows M=lane&15.
// C/D: VGPR v -> row M=v (lanes 0..15) / M=v+8 (lanes 16..31), N=lane&15.
// ---------------------------------------------------------------------------
__global__ __launch_bounds__(32 * TILES_PER_BLOCK)
void rgcn_xr_gemm(const float* __restrict__ feat,
                  const float* __restrict__ Wp,
                  float* __restrict__ xr, int N) {
  const int r    = blockIdx.y;
  const int wave = threadIdx.x >> 5;
  const int row0 = (blockIdx.x * TILES_PER_BLOCK + wave) * 16;
  if (row0 >= N) return;                 // whole wave exits; EXEC stays full
  const int lane = threadIdx.x & 31;
  const int half = lane >> 4;            // 0: K lo pair, 1: K hi pair
  const int l15  = lane & 15;
  const float* Wr = Wp + (size_t)r * D * D;

  int ar = row0 + l15; if (ar >= N) ar = N - 1;        // clamp (stores guarded)
  const float* arow = feat + (size_t)ar * D + half * 2;

  v2f a[16];
#pragma unroll
  for (int kk = 0; kk < 16; ++kk)
    a[kk] = *(const v2f*)(arow + kk * 4);

  const bool full = (row0 + 16 <= N);    // wave-uniform fast path
#pragma unroll
  for (int ct = 0; ct < 4; ++ct) {
    v8f c = {0.f, 0.f, 0.f, 0.f, 0.f, 0.f, 0.f, 0.f};
    const int col = ct * 16 + l15;
    const float* Bp = Wr + col * 2;      // + pair*2*D per k-step
#pragma unroll
    for (int kk = 0; kk < 16; ++kk) {
      const int pair = kk * 2 + half;    // K-pair index 0..31
      v2f b = *(const v2f*)(Bp + pair * 2 * D);
      c = wmma_f32_16x16x4(a[kk], b, c);
    }
    float* obase = xr + ((size_t)r * N + row0 + half * 8) * D + col;
    if (full) {
#pragma unroll
      for (int v = 0; v < 8; ++v)
        obase[(size_t)v * D] = c[v];
    } else {
#pragma unroll
      for (int v = 0; v < 8; ++v)
        if (row0 + half * 8 + v < N) obase[(size_t)v * D] = c[v];
    }
  }
}

// ---------------------------------------------------------------------------
// K3: per-edge gather * norm, scatter-add into agg (= d_out, pre-zeroed).
// One wave per edge; lane handles 2 consecutive floats (coalesced 256B/wave).
// Edge id is wave-uniform -> src/dst/etype/norm scalarize to SMEM loads.
// ---------------------------------------------------------------------------
__global__ __launch_bounds__(256)
void rgcn_edge_scatter(const float* __restrict__ xr,
                       const int* __restrict__ src,
                       const int* __restrict__ dst,
                       const int* __restrict__ etype,
                       const float* __restrict__ norm,
                       float* __restrict__ agg, int E, int N) {
  const int e = blockIdx.x * (256 / 32) + (threadIdx.x >> 5);
  if (e >= E) return;
  const int lane = threadIdx.x & 31;
  const int s = src[e];
  const int d = dst[e];
  const int r = etype[e];
  const float nv = norm[e];

  const float* row = xr + ((size_t)r * N + s) * D + lane * 2;
  v2f v = *(const v2f*)row;
  float* o = agg + (size_t)d * D + lane * 2;
  __hip_atomic_fetch_add(o + 0, v.x * nv, __ATOMIC_RELAXED, __HIP_MEMORY_SCOPE_AGENT);
  __hip_atomic_fetch_add(o + 1, v.y * nv, __ATOMIC_RELAXED, __HIP_MEMORY_SCOPE_AGENT);
}

// ---------------------------------------------------------------------------
// K4: out = relu(feat @ self_loop_weight + out)   (out holds agg sums)
// ---------------------------------------------------------------------------
__global__ __launch_bounds__(32 * TILES_PER_BLOCK)
void rgcn_selfloop_relu(const float* __restrict__ feat,
                        const float* __restrict__ Wp,
                        float* __restrict__ out, int N) {
  const int wave = threadIdx.x >> 5;
  const int row0 = (blockIdx.x * TILES_PER_BLOCK + wave) * 16;
  if (row0 >= N) return;
  const int lane = threadIdx.x & 31;
  const int half = lane >> 4;
  const int l15  = lane & 15;

  int ar = row0 + l15; if (ar >= N) ar = N - 1;
  const float* arow = feat + (size_t)ar * D + half * 2;

  v2f a[16];
#pragma unroll
  for (int kk = 0; kk < 16; ++kk)
    a[kk] = *(const v2f*)(arow + kk * 4);

  const bool full = (row0 + 16 <= N);
#pragma unroll
  for (int ct = 0; ct < 4; ++ct) {
    v8f c = {0.f, 0.f, 0.f, 0.f, 0.f, 0.f, 0.f, 0.f};
    const int col = ct * 16 + l15;
    const float* Bp = Wp + col * 2;
#pragma unroll
    for (int kk = 0; kk < 16; ++kk) {
      const int pair = kk * 2 + half;
      v2f b = *(const v2f*)(Bp + pair * 2 * D);
      c = wmma_f32_16x16x4(a[kk], b, c);
    }
    float* obase = out + ((size_t)row0 + half * 8) * D + col;
    if (full) {
#pragma unroll
      for (int v = 0; v < 8; ++v) {
        const float t = c[v] + obase[(size_t)v * D];
        obase[(size_t)v * D] = t > 0.f ? t : 0.f;
      }
    } else {
#pragma unroll
      for (int v = 0; v < 8; ++v) {
        if (row0 + half * 8 + v < N) {
          const float t = c[v] + obase[(size_t)v * D];
          obase[(size_t)v * D] = t > 0.f ? t : 0.f;
        }
      }
    }
  }
}

// ---------------------------------------------------------------------------
extern "C" void kernel_launch(void* const* d_in, const int* in_sizes, int n_in,
                              void* d_out, int out_size, void* d_ws, size_t ws_size,
                              hipStream_t stream) {
  const float* feat   = (const float*)d_in[0];
  const int*   src    = (const int*)  d_in[1];
  const int*   dst    = (const int*)  d_in[2];
  const int*   etype  = (const int*)  d_in[3];
  const float* norm   = (const float*)d_in[4];
  const float* weight = (const float*)d_in[5];
  const float* w_comp = (const float*)d_in[6];
  const float* slw    = (const float*)d_in[7];
  float* out = (float*)d_out;

  const int N = in_sizes[0] / D;       // 50000
  const int E = in_sizes[1];           // 800000
  const int R = in_sizes[6] / BASES;   // 10

  // Workspace: Wp [R*64*64] | slw_pack [64*64] | xr [R*N*64] (~128.2 MB,
  // xr stays resident in the 192 MB L2 -> edge gather mostly hits L2).
  float* Wp       = (float*)d_ws;
  float* slw_pack = Wp + (size_t)R * D * D;
  float* xr       = slw_pack + (size_t)D * D;

  // agg accumulator lives in d_out; zero it first (capture-safe memset node).
  hipMemsetAsync(d_out, 0, (size_t)N * D * sizeof(float), stream);

  const int wtot = R * D * D;
  rgcn_basis_combine<<<(wtot + 255) / 256, 256, 0, stream>>>(weight, w_comp, Wp, R);
  rgcn_pack_w64<<<(D * D + 255) / 256, 256, 0, stream>>>(slw, slw_pack);

  const int ntiles = (N + 15) / 16;
  dim3 gx((ntiles + TILES_PER_BLOCK - 1) / TILES_PER_BLOCK, R);
  rgcn_xr_gemm<<<gx, 32 * TILES_PER_BLOCK, 0, stream>>>(feat, Wp, xr, N);

  rgcn_edge_scatter<<<(E + 7) / 8, 256, 0, stream>>>(xr, src, dst, etype, norm,
                                                     out, E, N);

  rgcn_selfloop_relu<<<(ntiles + TILES_PER_BLOCK - 1) / TILES_PER_BLOCK,
                       32 * TILES_PER_BLOCK, 0, stream>>>(feat, slw_pack, out, N);
}